// ScaledDotProductAttention_1537598292042
// MI455X (gfx1250) — compile-verified
//
#include <hip/hip_runtime.h>
#include <hip/hip_bf16.h>

typedef __attribute__((ext_vector_type(2))) float v2f;
typedef __attribute__((ext_vector_type(8))) float v8f;

#define Sn 1024
#define Dh 64
#define Hh 16
#define SPAD (Sn + 4)
#define QPAD (Dh + 4)

// one block = one (b, h, 16-row q tile); 256 threads = 8 waves (wave32)
__global__ __launch_bounds__(256) void attn_wmma_f32_kernel(
    const float* __restrict__ q, const float* __restrict__ k,
    const float* __restrict__ v, const int* __restrict__ mask,
    const float* __restrict__ adj, const float* __restrict__ dist,
    float* __restrict__ out, float* __restrict__ attn_out)
{
  const int qt = blockIdx.x;          // 0..63  (q tile of 16 rows)
  const int h  = blockIdx.y;          // 0..15
  const int b  = blockIdx.z;          // 0..3
  const int bh = b * Hh + h;
  const int tid  = threadIdx.x;
  const int lane = tid & 31;
  const int wave = tid >> 5;          // 0..7
  const int hl = lane & 15;           // lane within half-wave
  const int hi = lane >> 4;           // half-wave select (0: K={0,1}, 1: K={2,3})

  __shared__ float Qs[16][QPAD];      // scaled Q tile
  __shared__ float Ps[16][SPAD];      // scores -> probabilities
  __shared__ float red[16][8];
  __shared__ float rowmax[16];
  __shared__ float rowinv[16];

  // ---- load Q tile, pre-scaled by 1/TEMPERATURE ----
  const float* qbase = q + ((size_t)bh * Sn + (size_t)qt * 16) * Dh;
  for (int i = tid; i < 16 * Dh; i += 256)
    Qs[i >> 6][i & 63] = qbase[i] * 0.125f;
  __syncthreads();

  const float* kb = k + (size_t)bh * Sn * Dh;
  const float* distrow0 = dist + ((size_t)b * Sn + (size_t)qt * 16) * Sn;
  const float* adjrow0  = adj  + ((size_t)b * Sn + (size_t)qt * 16) * Sn;

  // ---- phase 1: S tile = Q * K^T (WMMA f32 16x16x4), epilogue into LDS ----
  for (int jt = 0; jt < 8; ++jt) {
    const int j  = jt * 8 + wave;     // column tile 0..63
    const int n0 = j * 16;
    v8f acc = {};
    #pragma unroll
    for (int kk = 0; kk < 16; ++kk) {
      const int d0 = kk * 4 + hi * 2;
      v2f a;  a.x = Qs[hl][d0];  a.y = Qs[hl][d0 + 1];   // A: 16x4, M=hl
      const float* krow = kb + (size_t)(n0 + hl) * Dh + d0;
      v2f bb; bb.x = krow[0];    bb.y = krow[1];          // B: 4x16, N=hl
      acc = __builtin_amdgcn_wmma_f32_16x16x4_f32(
          false, a, false, bb, (short)0, acc, false, false);
    }
    // epilogue: relu * rescale(dist) + adj, mask -> -1e9
    const int mk = mask[b * Sn + n0 + hl];
    #pragma unroll
    for (int r = 0; r < 8; ++r) {
      const int m = hi * 8 + r;                 // C/D layout: lanes16-31 -> M+8
      const int n = hl;
      float s  = acc[r];
      float dd = distrow0[(size_t)m * Sn + n0 + n];
      // (1+e)/(1+e^(1-d)) via single v_rcp_f32 instead of IEEE divide expansion
      float resc = 3.7182818284590452f *
                   __builtin_amdgcn_rcpf(1.0f + __expf(1.0f - dd));
      s = fmaxf(s, 0.0f) * resc + adjrow0[(size_t)m * Sn + n0 + n];
      if (mk == 0) s = -1.0e9f;
      Ps[m][n0 + n] = s;
    }
  }
  __syncthreads();

  // ---- softmax over the 16x1024 strip ----
  if (tid < 128) {                    // 8 partials per row
    const int row = tid >> 3, part = tid & 7;
    float mx = -3.0e38f;
    for (int n = part * 128; n < part * 128 + 128; ++n) mx = fmaxf(mx, Ps[row][n]);
    red[row][part] = mx;
  }
  __syncthreads();
  if (tid < 16) {
    float mx = red[tid][0];
    #pragma unroll
    for (int p = 1; p < 8; ++p) mx = fmaxf(mx, red[tid][p]);
    rowmax[tid] = mx;
  }
  __syncthreads();
  if (tid < 128) {
    const int row = tid >> 3, part = tid & 7;
    const float mx = rowmax[row];
    float sum = 0.0f;
    for (int n = part * 128; n < part * 128 + 128; ++n) {
      float e = __expf(Ps[row][n] - mx);
      Ps[row][n] = e;
      sum += e;
    }
    red[row][part] = sum;
  }
  __syncthreads();
  if (tid < 16) {
    float s = 0.0f;
    #pragma unroll
    for (int p = 0; p < 8; ++p) s += red[tid][p];
    rowinv[tid] = __builtin_amdgcn_rcpf(s);
  }
  __syncthreads();

  // ---- normalize in LDS and stream attention matrix to global (coalesced) ----
  float* attn_base = attn_out + ((size_t)bh * Sn + (size_t)qt * 16) * Sn;
  for (int i = tid; i < 16 * Sn; i += 256) {
    const int m = i >> 10, n = i & (Sn - 1);
    const float p = Ps[m][n] * rowinv[m];
    Ps[m][n] = p;
    attn_base[(size_t)m * Sn + n] = p;
  }
  __syncthreads();

  // ---- phase 2: out tile = P * V (16x64), waves 0..3 each own 16 D-columns ----
  if (wave < 4) {
    const int n0 = wave * 16;
    const float* vb = v + (size_t)bh * Sn * Dh;
    v8f acc = {};
    for (int kk = 0; kk < 256; ++kk) {
      const int k0 = kk * 4 + hi * 2;
      v2f a;  a.x = Ps[hl][k0];  a.y = Ps[hl][k0 + 1];   // A: 16x4 from probs
      const float* vrow = vb + (size_t)k0 * Dh + n0 + hl;
      v2f bb; bb.x = vrow[0];    bb.y = vrow[Dh];         // B: 4x16 from V
      acc = __builtin_amdgcn_wmma_f32_16x16x4_f32(
          false, a, false, bb, (short)0, acc, false, false);
    }
    float* ob = out + ((size_t)bh * Sn + (size_t)qt * 16) * Dh + n0;
    #pragma unroll
    for (int r = 0; r < 8; ++r) {
      const int m = hi * 8 + r;
      ob[(size_t)m * Dh + hl] = acc[r];
    }
  }
}

extern "C" void kernel_launch(void* const* d_in, const int* in_sizes, int n_in,
                              void* d_out, int out_size, void* d_ws, size_t ws_size,
                              hipStream_t stream) {
  (void)in_sizes; (void)n_in; (void)d_ws; (void)ws_size; (void)out_size;
  const float* q    = (const float*)d_in[0];
  const float* k    = (const float*)d_in[1];
  const float* v    = (const float*)d_in[2];
  const int*   mask = (const int*)d_in[3];
  const float* adj  = (const float*)d_in[4];
  const float* dist = (const float*)d_in[5];

  float* out_base  = (float*)d_out;                       // (B,H,S,D) first
  float* attn_base = out_base + (size_t)4 * Hh * Sn * Dh; // then (B,H,S,S)

  dim3 grid(Sn / 16, Hh, 4);   // 64 x 16 x 4 = 4096 blocks
  dim3 block(256);
  attn_wmma_f32_kernel<<<grid, block, 0, stream>>>(q, k, v, mask, adj, dist,
                                                   out_base, attn_base);
}